// SparseConvolutionDownsample_33638183862317
// MI455X (gfx1250) — compile-verified
//
#include <hip/hip_runtime.h>
#include <hip/hip_bf16.h>

// ---------------------------------------------------------------------------
// Sparse conv downsample + BN + ReLU for MI455X (gfx1250).
//
// y[m] = z[m] @ Wcat,  Wcat = concat(W[0..3]) : [128,64],
// z[m] (len 128) = x[2m] placed at k_idx[2m]*32  +  x[2m+1] at k_idx[2m+1]*32.
// Uniform-B trick -> V_WMMA_F32_16X16X4_F32 (full fp32 precision).
// Roofline: ~768MB @ 23.3TB/s ~= 33us -> HBM-bound; two recompute GEMM passes
// (stats, then normalize) avoid a 256MB y spill.
// x tiles staged via GLOBAL_LOAD_ASYNC_TO_LDS_B64 (ASYNCcnt), double-buffered
// so the 32KB/chunk stage overlaps the 1024-WMMA/chunk compute.
// ---------------------------------------------------------------------------

#define MOUT     1048576
#define CIN      32
#define COUT     64
#define KD       128         // 4*CIN expanded K
#define CHUNK_M  128         // outputs per chunk = 8 waves * 16
#define CHUNKS   8           // chunks per block
#define NBLOCKS  1024        // MOUT / (CHUNK_M*CHUNKS)
#define WSTR     80          // W row stride (dwords): B-frag's 64 dwords hit all
                             // 64 banks exactly once (k+1 -> +16, +2 rows -> +32)
#define XSTR     34          // x row stride (dwords): 8B-aligned b64 LDS dests,
                             // 32 rows -> 32 distinct even bank bases
#define XROWS    256         // 2*CHUNK_M staged input rows per chunk
#define BN_EPS   1e-4f

typedef float v2f __attribute__((ext_vector_type(2)));
typedef float v8f __attribute__((ext_vector_type(8)));

// ---- gfx1250 async global->LDS path (guarded; host pass takes fallback) ----
#if defined(__gfx1250__) && __has_builtin(__builtin_amdgcn_global_load_async_to_lds_b64)
#define HAVE_ASYNC_LDS 1
#else
#define HAVE_ASYNC_LDS 0
#endif

// Builtin parameter types (per hipcc diagnostic): GCC-style <2 x i32> pointers
// in addrspace(1) (global) and addrspace(3) (LDS).
typedef int v2i_b64 __attribute__((vector_size(8)));
#define AS1 __attribute__((address_space(1)))
#define AS3 __attribute__((address_space(3)))

__device__ __forceinline__ void async_copy_b64(float* lds_dst, const float* gsrc) {
#if HAVE_ASYNC_LDS
  __builtin_amdgcn_global_load_async_to_lds_b64(
      (AS1 v2i_b64*)gsrc, (AS3 v2i_b64*)lds_dst, /*offset=*/0, /*cpol=*/0);
#else
  lds_dst[0] = gsrc[0];
  lds_dst[1] = gsrc[1];
#endif
}

#if defined(__gfx1250__)
#if __has_builtin(__builtin_amdgcn_s_wait_asynccnt)
#define WAIT_ASYNC(n) __builtin_amdgcn_s_wait_asynccnt(n)
#else
#define WAIT_ASYNC(n) asm volatile("s_wait_asynccnt %0" ::"n"(n) : "memory")
#endif
#else
#define WAIT_ASYNC(n) do {} while (0)
#endif

template <int PASS>
__device__ __forceinline__ void sparse_conv_pass(
    const float* __restrict__ x, const float* __restrict__ W,
    const int* __restrict__ k_idx, float* __restrict__ out,
    float* __restrict__ ws)
{
  __shared__ float Wlds[KD * WSTR];          // 40.0 KB
  __shared__ float Xlds[2][XROWS * XSTR];    // 2 x 34.0 KB (double buffer)
  __shared__ float s_sum[COUT];
  __shared__ float s_sq[COUT];

  const int tid  = threadIdx.x;
  const int lane = tid & 31;
  const int wave = tid >> 5;
  const int l15  = lane & 15;
  const int hh   = lane >> 4;          // half-wave: K-pair select in A/B frags

  // Stage Wcat = W flattened [128][64] into padded LDS (one-time, L2-resident).
  for (int i = tid; i < KD * COUT; i += 256)
    Wlds[(i >> 6) * WSTR + (i & 63)] = W[i];

  float sc[4], bi[4];
  if (PASS == 1) {
    if (tid < COUT) { s_sum[tid] = 0.0f; s_sq[tid] = 0.0f; }
  } else {
    #pragma unroll
    for (int n = 0; n < 4; ++n) {
      const int c = 16 * n + l15;
      sc[n] = ws[128 + c];
      bi[n] = ws[192 + c];
    }
  }

  // Issue async stage of chunk `nc`'s 256 x-rows into buffer nc&1.
  // 16 b64 ops per thread -> ASYNCcnt rises by 16 per wave per chunk.
  auto issue_stage = [&](int nc) {
    const size_t gbase = (size_t)((blockIdx.x * CHUNKS + nc) * CHUNK_M * 2) * CIN;
    float* buf = Xlds[nc & 1];
    #pragma unroll
    for (int it = 0; it < 16; ++it) {
      const int g   = it * 512 + tid * 2;   // dword index in 8192-dword tile
      const int row = g >> 5, col = g & 31; // col is even -> 8B aligned
      async_copy_b64(&buf[row * XSTR + col], &x[gbase + g]);
    }
  };

  issue_stage(0);

  for (int ch = 0; ch < CHUNKS; ++ch) {
    // Prefetch next chunk, then wait only for *this* chunk's copies
    // (in-order completion: <=16 outstanding means the older 16 retired).
    if (ch + 1 < CHUNKS) { issue_stage(ch + 1); WAIT_ASYNC(16); }
    else                 { WAIT_ASYNC(0); }
    __syncthreads();                         // cross-wave LDS visibility

    const float* Xb = Xlds[ch & 1];
    const int out_base = (blockIdx.x * CHUNKS + ch) * CHUNK_M;

    const int lo  = wave * 16 + l15;         // A-matrix row M = lane&15
    const int gm  = out_base + lo;
    const int kk0 = k_idx[2 * gm];
    const int kk1 = k_idx[2 * gm + 1];
    const int lr0 = 2 * lo;
    const int lr1 = lr0 + 1;

    v8f acc[4];
    #pragma unroll
    for (int n = 0; n < 4; ++n)
      #pragma unroll
      for (int r = 0; r < 8; ++r) acc[n][r] = 0.0f;

    // K=128 -> 32 chained f32 WMMAs per 16x16 tile, 4 N-tiles.
    for (int s = 0; s < 32; ++s) {
      const int koff = s >> 3;                       // which 32-wide W chunk
      const int c0   = ((s & 7) << 2) + (hh << 1);   // col within chunk
      const bool f0  = (kk0 == koff);
      const bool f1  = (kk1 == koff);
      const int  lr  = f0 ? lr0 : lr1;               // v_cndmask, EXEC stays full
      const float mk = (f0 || f1) ? 1.0f : 0.0f;
      v2f a;
      a.x = mk * Xb[lr * XSTR + c0];
      a.y = mk * Xb[lr * XSTR + c0 + 1];

      const int k0 = (s << 2) + (hh << 1);           // lanes16-31 hold K+2,K+3
      #pragma unroll
      for (int n = 0; n < 4; ++n) {
        v2f b;
        b.x = Wlds[k0 * WSTR + 16 * n + l15];
        b.y = Wlds[(k0 + 1) * WSTR + 16 * n + l15];
        acc[n] = __builtin_amdgcn_wmma_f32_16x16x4_f32(
            false, a, false, b, (short)0, acc[n], false, false);
      }
    }

    if (PASS == 1) {
      // Per-channel batch stats. C/D layout: lane l, VGPR r -> y[r + hh*8][l15].
      #pragma unroll
      for (int n = 0; n < 4; ++n) {
        float ps = 0.0f, pq = 0.0f;
        #pragma unroll
        for (int r = 0; r < 8; ++r) { const float v = acc[n][r]; ps += v; pq += v * v; }
        const int c = 16 * n + l15;
        atomicAdd(&s_sum[c], ps);
        atomicAdd(&s_sq[c], pq);
      }
    } else {
      const int row_base = out_base + wave * 16 + hh * 8;
      #pragma unroll
      for (int n = 0; n < 4; ++n) {
        const int c = 16 * n + l15;
        #pragma unroll
        for (int r = 0; r < 8; ++r) {
          const float v = fmaxf(acc[n][r] * sc[n] + bi[n], 0.0f);
          __builtin_nontemporal_store(v, &out[(size_t)(row_base + r) * COUT + c]);
        }
      }
    }
    __syncthreads();   // all reads of buf[ch&1] done before it is re-staged
  }

  if (PASS == 1) {
    if (tid < COUT) {
      atomicAdd(&ws[tid],        s_sum[tid]);
      atomicAdd(&ws[COUT + tid], s_sq[tid]);
    }
  }
}

__global__ __launch_bounds__(256) void sc_pass1_kernel(
    const float* __restrict__ x, const float* __restrict__ W,
    const int* __restrict__ k_idx, float* __restrict__ ws)
{
  sparse_conv_pass<1>(x, W, k_idx, nullptr, ws);
}

__global__ __launch_bounds__(256) void sc_pass2_kernel(
    const float* __restrict__ x, const float* __restrict__ W,
    const int* __restrict__ k_idx, float* __restrict__ out,
    float* __restrict__ ws)
{
  sparse_conv_pass<2>(x, W, k_idx, out, ws);
}

__global__ void bn_finalize_kernel(const float* __restrict__ gamma,
                                   const float* __restrict__ beta,
                                   float* __restrict__ ws)
{
  const int c = threadIdx.x;
  if (c < COUT) {
    const float mean = ws[c] * (1.0f / MOUT);
    const float var  = ws[COUT + c] * (1.0f / MOUT) - mean * mean;
    const float s    = gamma[c] * rsqrtf(var + BN_EPS);
    ws[128 + c] = s;
    ws[192 + c] = beta[c] - mean * s;
  }
}

extern "C" void kernel_launch(void* const* d_in, const int* in_sizes, int n_in,
                              void* d_out, int out_size, void* d_ws, size_t ws_size,
                              hipStream_t stream)
{
  const float* x     = (const float*)d_in[0];   // [N, 32]
  const float* W     = (const float*)d_in[1];   // [4, 32, 64] == Wcat [128,64]
  const float* gamma = (const float*)d_in[2];   // [64]
  const float* beta  = (const float*)d_in[3];   // [64]
  const int*   k_idx = (const int*)d_in[5];     // [N]  (out_idx == i/2 by construction)
  float*       out   = (float*)d_out;           // [M, 64]
  float*       ws    = (float*)d_ws;            // 256 floats: sum|sumsq|scale|bias

  (void)hipMemsetAsync(ws, 0, 2 * COUT * sizeof(float), stream);  // graph-capturable
  hipLaunchKernelGGL(sc_pass1_kernel, dim3(NBLOCKS), dim3(256), 0, stream,
                     x, W, k_idx, ws);
  hipLaunchKernelGGL(bn_finalize_kernel, dim3(1), dim3(64), 0, stream,
                     gamma, beta, ws);
  hipLaunchKernelGGL(sc_pass2_kernel, dim3(NBLOCKS), dim3(256), 0, stream,
                     x, W, k_idx, out, ws);
}